// Pooling_25950192403296
// MI455X (gfx1250) — compile-verified
//
#include <hip/hip_runtime.h>
#include <hip/hip_bf16.h>

typedef __attribute__((ext_vector_type(2))) float v2f;
typedef __attribute__((ext_vector_type(8))) float v8f;

#define NA    1024
#define HID   128
#define GS    32          // S: grid side
#define NCELL (GS * GS)   // 1024 cells
#define KDIM  2048        // HID * 4 * 4
#define ODIM  128

// ---------------------------------------------------------------------------
// Kernel 1: social pooling. One block per ego agent.
//  Phase A: pairwise cell assignment with per-cell winner (atomicMax dedupe,
//           deterministic stand-in for jnp .set collision semantics).
//  Phase B: compact occupied cells into a list (LDS counter).
//  Phase C: ds_add_f32-accumulate hidden[winner] into 16x128 pooled tile.
//  Phase D: emit channel-first flat row [h*16 + g0*4 + g1] to workspace.
// ---------------------------------------------------------------------------
__global__ __launch_bounds__(256) void social_pool_kernel(
    const float* __restrict__ hidden,   // [NA, HID]
    const float* __restrict__ obs2,     // [NA, 2]
    float* __restrict__ flatA)          // [NA, KDIM] workspace
{
  __shared__ int   winner[NCELL];
  __shared__ int   list[NCELL];
  __shared__ float pooled[16 * HID];
  __shared__ int   count;

  const int i   = blockIdx.x;
  const int tid = threadIdx.x;

  for (int c = tid; c < NCELL; c += 256) winner[c] = -1;
  for (int t = tid; t < 16 * HID; t += 256) pooled[t] = 0.0f;
  if (tid == 0) count = 0;
  __syncthreads();

  const float ex = obs2[2 * i];
  const float ey = obs2[2 * i + 1];
  const bool ego_ok = (ex == ex);   // ~isnan

  if (ego_ok) {
    for (int j = tid; j < NA; j += 256) {
      const float jx = obs2[2 * j];
      const float jy = obs2[2 * j + 1];
      const bool jok = (jx == jx);
      // oij = rel / (CELL_SIDE/POOL) + S/2 = rel*4 + 16
      const float o0 = (jx - ex) * 4.0f + 16.0f;
      const float o1 = (jy - ey) * 4.0f + 16.0f;
      const bool inr = (o0 >= 0.0f) && (o0 < 32.0f) &&
                       (o1 >= 0.0f) && (o1 < 32.0f);
      if (inr && jok && (j != i)) {
        const int c0 = (int)o0;
        const int c1 = (int)o1;
        atomicMax(&winner[c0 * GS + c1], j);
      }
    }
  }
  __syncthreads();

  for (int c = tid; c < NCELL; c += 256) {
    if (winner[c] >= 0) {
      const int p = atomicAdd(&count, 1);
      list[p] = c;
    }
  }
  __syncthreads();

  const int cnt  = count;
  const int h    = tid & (HID - 1);
  const int sub  = tid >> 7;              // two cells in flight per step
  for (int k = sub; k < cnt; k += 2) {
    const int c  = list[k];
    const int w  = winner[c];
    const int c0 = c >> 5, c1 = c & 31;
    const int g  = (c0 >> 3) * 4 + (c1 >> 3);   // 4x4 pooled block
    atomicAdd(&pooled[g * HID + h], hidden[w * HID + h]);  // ds_add_f32
  }
  __syncthreads();

  float* dst = flatA + (size_t)i * KDIM;
  #pragma unroll
  for (int t = 0; t < 8; ++t) {
    const int fi = tid + 256 * t;       // flat = h*16 + g
    const int hh = fi >> 4;
    const int g  = fi & 15;
    dst[fi] = pooled[g * HID + hh];
  }
}

// ---------------------------------------------------------------------------
// Kernel 2: out = relu(flat @ W^T + b) with V_WMMA_F32_16X16X4_F32.
// grid = 64 M-tiles; 8 waves/block, wave w owns N-columns [16w, 16w+16).
// A frag (16x4): lane r=lane&15 -> row M=r; half=lane>>4 selects K pair
//   {0,1} or {2,3}  => contiguous float2 load from A row.
// B frag (4x16): B[k][n] = W[n][k]   => contiguous float2 load from W row.
// C/D: VGPR t, lane -> (M = t + 8*half, N = r).
// ---------------------------------------------------------------------------
__global__ __launch_bounds__(256) void gemm_relu_wmma(
    const float* __restrict__ A,      // [NA, KDIM]
    const float* __restrict__ Wm,     // [ODIM, KDIM]
    const float* __restrict__ bias,   // [ODIM]
    float* __restrict__ out)          // [NA, ODIM]
{
  const int lane  = threadIdx.x & 31;
  const int wave  = threadIdx.x >> 5;
  const int r     = lane & 15;
  const int half  = lane >> 4;
  const int mtile = blockIdx.x * 16;
  const int ntile = wave * 16;

  const float* arow = A  + (size_t)(mtile + r) * KDIM + 2 * half;
  const float* wrow = Wm + (size_t)(ntile + r) * KDIM + 2 * half;

  v8f c = {};
  #pragma unroll 8
  for (int kk = 0; kk < KDIM; kk += 4) {
    const v2f a = *(const v2f*)(arow + kk);
    const v2f b = *(const v2f*)(wrow + kk);
    // 8 args: (neg_a, A, neg_b, B, c_mod, C, reuse_a, reuse_b)
    c = __builtin_amdgcn_wmma_f32_16x16x4_f32(
        false, a, false, b, (short)0, c, false, false);
  }

  const float bv = bias[ntile + r];
  #pragma unroll
  for (int t = 0; t < 8; ++t) {
    const float v = c[t] + bv;
    out[(size_t)(mtile + t + 8 * half) * ODIM + (ntile + r)] =
        v > 0.0f ? v : 0.0f;
  }
}

// ---------------------------------------------------------------------------
extern "C" void kernel_launch(void* const* d_in, const int* in_sizes, int n_in,
                              void* d_out, int out_size, void* d_ws, size_t ws_size,
                              hipStream_t stream) {
  // setup_inputs order: hidden_state, obs1, obs2, W, b   (obs1 unused)
  const float* hidden = (const float*)d_in[0];
  const float* obs2   = (const float*)d_in[2];
  const float* Wm     = (const float*)d_in[3];
  const float* bias   = (const float*)d_in[4];

  float* flatA = (float*)d_ws;   // [1024, 2048] fp32 = 8 MB scratch

  social_pool_kernel<<<NA, 256, 0, stream>>>(hidden, obs2, flatA);
  gemm_relu_wmma<<<NA / 16, 256, 0, stream>>>(flatA, Wm, bias, (float*)d_out);
}